// SimpleDotAttention_5231270166738
// MI455X (gfx1250) — compile-verified
//
#include <hip/hip_runtime.h>
#include <hip/hip_bf16.h>

// Problem constants from the reference: q,k are [E, M=2, H=8, C=8] f32,
// index sorted int, N nodes. Output [E, H] f32.
#define H 8
#define QK 128              // M*H*C floats per edge
#define TPB 256             // 8 waves of 32 on CDNA5
#define EPT1 2              // edges per thread in pass 1
#define EPB (TPB / H * EPT1)        // 64 edges per block
#define TILE_ELEMS (EPB * QK)       // 8192 floats per matrix per block (32 KB)
#define EPT2 8              // edges per thread in pass 2 (atomic aggregation)

#define SCALE 0.35355339059327373f  // 8^-0.5
#define NEG_INF (-__builtin_inff())

typedef unsigned int u32x4 __attribute__((ext_vector_type(4)));
typedef int          i32x8 __attribute__((ext_vector_type(8)));
typedef int          i32x4 __attribute__((ext_vector_type(4)));

#if __has_builtin(__builtin_amdgcn_tensor_load_to_lds)
#define HAVE_TDM 1
#else
#define HAVE_TDM 0
#endif

__device__ inline void wait_tensorcnt0() {
#if __has_builtin(__builtin_amdgcn_s_wait_tensorcnt)
    __builtin_amdgcn_s_wait_tensorcnt(0);
#else
    asm volatile("s_wait_tensorcnt 0x0" ::: "memory");
#endif
}

#if HAVE_TDM
// 1-D TDM descriptor: move `nelem` contiguous f32 from global to LDS.
// D# packing per CDNA5 ISA ch.8 (group0 128b, group1 256b; groups 2/3 zero
// for <=2D tensors). data_size=4B, tile_dim0=tensor_dim0=nelem, dim1=1.
// This toolchain (clang-23/therock) exposes the 6-arg builtin form.
__device__ inline void tdm_load_1d(const float* gsrc, float* lds_dst, unsigned nelem) {
    unsigned long long ga = (unsigned long long)(uintptr_t)gsrc;
    u32x4 g0;
    g0.x = 1u;                                    // count=1, is_restore=0, gather=0
    g0.y = (unsigned)(uintptr_t)lds_dst;          // LDS byte address (flat addr low 32)
    g0.z = (unsigned)ga;                          // global_addr[31:0]
    g0.w = ((unsigned)(ga >> 32) & 0x01FFFFFFu)   // global_addr[56:32]
         | (2u << 30);                            // type=2 ("image")
    i32x8 g1;
    g1[0] = (int)(2u << 16);                      // workgroup_mask=0, data_size=2 (4B)
    g1[1] = (int)((nelem & 0xFFFFu) << 16);       // tensor_dim0[15:0]
    g1[2] = (int)((nelem >> 16) | (1u << 16));    // tensor_dim0[31:16] | tensor_dim1=1
    g1[3] = (int)((nelem & 0xFFFFu) << 16);       // tile_dim0 = nelem (<= 8192)
    g1[4] = 0;                                    // tile_dim1=0 (unused), tile_dim2=0
    g1[5] = (int)nelem;                           // tensor_dim0_stride[31:0]
    g1[6] = 0;
    g1[7] = 0;
    i32x4 z4 = {0, 0, 0, 0};
    i32x8 z8 = {0, 0, 0, 0, 0, 0, 0, 0};
    __builtin_amdgcn_tensor_load_to_lds(g0, g1, z4, z4, z8, 0);
}
#endif

__device__ inline void atomicMaxF(float* addr, float val) {
    unsigned* ua = (unsigned*)addr;
    unsigned cur = *ua;
    while (val > __uint_as_float(cur)) {
        unsigned prev = atomicCAS(ua, cur, __float_as_uint(val));
        if (prev == cur) break;
        cur = prev;
    }
}

// ---------------------------------------------------------------- init ----
__global__ void sda_init(float* seg_max, float* seg_sum, const int* nnodes) {
    int nh = nnodes[0] * H;
    for (int i = blockIdx.x * blockDim.x + threadIdx.x; i < nh;
         i += gridDim.x * blockDim.x) {
        seg_max[i] = NEG_INF;
        seg_sum[i] = 0.0f;
    }
}

// ---------------------------------------------------- pass 1: pre + max ----
__global__ void __launch_bounds__(TPB)
sda_pass1(const float* __restrict__ q, const float* __restrict__ k,
          const int* __restrict__ index, float* __restrict__ pre_out,
          float* __restrict__ seg_max, int E) {
    __shared__ __align__(16) float sq[TILE_ELEMS];
    __shared__ __align__(16) float sk[TILE_ELEMS];

    const long long e0 = (long long)blockIdx.x * EPB;
    const unsigned elems =
        (unsigned)(((long long)E - e0) * QK < (long long)TILE_ELEMS
                       ? ((long long)E - e0) * QK
                       : (long long)TILE_ELEMS);

#if HAVE_TDM
    if (threadIdx.x < 32) {            // wave 0 only issues the DMA descriptors
        tdm_load_1d(q + e0 * QK, sq, elems);
        tdm_load_1d(k + e0 * QK, sk, elems);
        wait_tensorcnt0();
    }
    __syncthreads();
#else
    {
        const float4* q4 = (const float4*)(q + e0 * QK);
        const float4* k4 = (const float4*)(k + e0 * QK);
        float4* sq4 = (float4*)sq;
        float4* sk4 = (float4*)sk;
        unsigned n4 = elems / 4;
        for (unsigned i = threadIdx.x; i < n4; i += TPB) {
            sq4[i] = q4[i];
            sk4[i] = k4[i];
        }
        __syncthreads();
    }
#endif

    const int h = threadIdx.x & (H - 1);
    const int s = threadIdx.x >> 3;             // 0..31
    const float4* sq4 = (const float4*)sq;
    const float4* sk4 = (const float4*)sk;

    float rmax = NEG_INF;
    int   rnode = -1;
    for (int j = 0; j < EPT1; ++j) {
        const int el = s * EPT1 + j;
        const long long e = e0 + el;
        if (e >= E) break;
        // per (edge, h): floats m*64 + h*8 .. +7  ->  float4 idx m*16 + h*2
        const int b = el * 32 + h * 2;
        float4 a0 = sq4[b +  0], a1 = sq4[b +  1];   // m=0
        float4 a2 = sq4[b + 16], a3 = sq4[b + 17];   // m=1
        float4 b0 = sk4[b +  0], b1 = sk4[b +  1];
        float4 b2 = sk4[b + 16], b3 = sk4[b + 17];
        float dot = a0.x*b0.x + a0.y*b0.y + a0.z*b0.z + a0.w*b0.w
                  + a1.x*b1.x + a1.y*b1.y + a1.z*b1.z + a1.w*b1.w
                  + a2.x*b2.x + a2.y*b2.y + a2.z*b2.z + a2.w*b2.w
                  + a3.x*b3.x + a3.y*b3.y + a3.z*b3.z + a3.w*b3.w;
        const float pre = dot * SCALE;
        pre_out[e * H + h] = pre;

        const int node = index[e];
        if (node != rnode) {               // index is sorted: aggregate runs
            if (rnode >= 0) atomicMaxF(&seg_max[rnode * H + h], rmax);
            rnode = node;
            rmax = pre;
        } else {
            rmax = fmaxf(rmax, pre);
        }
    }
    if (rnode >= 0) atomicMaxF(&seg_max[rnode * H + h], rmax);
}

// ----------------------------------------------------- pass 2: exp + sum ----
__global__ void __launch_bounds__(TPB)
sda_pass2(const int* __restrict__ index, float* __restrict__ ex_inout,
          const float* __restrict__ seg_max, float* __restrict__ seg_sum, int E) {
    const long long t = (long long)blockIdx.x * TPB + threadIdx.x;
    const int h = (int)(t & (H - 1));
    const long long s = t >> 3;
    const long long ebase = s * EPT2;
    if (ebase >= E) return;

    float run = 0.0f;
    int   rnode = -1;
    float m = 0.0f;
    for (int j = 0; j < EPT2; ++j) {
        const long long e = ebase + j;
        if (e >= E) break;
        const int node = index[e];
        if (node != rnode) {
            if (rnode >= 0) atomicAdd(&seg_sum[rnode * H + h], run);
            rnode = node;
            run = 0.0f;
            m = seg_max[node * H + h];
        }
        const float ex = __expf(ex_inout[e * H + h] - m);
        ex_inout[e * H + h] = ex;
        run += ex;
    }
    if (rnode >= 0) atomicAdd(&seg_sum[rnode * H + h], run);
}

// ------------------------------------------------------ pass 3: normalize ----
__global__ void __launch_bounds__(TPB)
sda_pass3(const int* __restrict__ index, float* __restrict__ out,
          const float* __restrict__ seg_sum, long long total) {
    const long long i = (long long)blockIdx.x * TPB + threadIdx.x;
    if (i >= total) return;
    const long long e = i >> 3;
    const int h = (int)(i & (H - 1));
    out[i] = out[i] / (seg_sum[(long long)index[e] * H + h] + 1e-16f);
}

// ------------------------------------------------------------- launcher ----
extern "C" void kernel_launch(void* const* d_in, const int* in_sizes, int n_in,
                              void* d_out, int out_size, void* d_ws, size_t ws_size,
                              hipStream_t stream) {
    const float* q     = (const float*)d_in[0];
    const float* k     = (const float*)d_in[1];
    const int*   index = (const int*)d_in[2];
    const int*   nn    = (const int*)d_in[3];
    float*       out   = (float*)d_out;

    const int E = in_sizes[0] / QK;

    // N is device-only; split workspace in half: [seg_max | seg_sum].
    float* seg_max = (float*)d_ws;
    size_t half = (ws_size / 2) & ~(size_t)255;
    float* seg_sum = (float*)((char*)d_ws + half);

    sda_init<<<512, TPB, 0, stream>>>(seg_max, seg_sum, nn);

    const int blocks1 = (E + EPB - 1) / EPB;
    sda_pass1<<<blocks1, TPB, 0, stream>>>(q, k, index, out, seg_max, E);

    const long long slots = ((long long)E + EPT2 - 1) / EPT2;
    const long long thr2 = slots * H;
    const int blocks2 = (int)((thr2 + TPB - 1) / TPB);
    sda_pass2<<<blocks2, TPB, 0, stream>>>(index, out, seg_max, seg_sum, E);

    const long long total = (long long)E * H;
    const int blocks3 = (int)((total + TPB - 1) / TPB);
    sda_pass3<<<blocks3, TPB, 0, stream>>>(index, out, seg_sum, total);
}